// GCN_51084341018872
// MI455X (gfx1250) — compile-verified
//
#include <hip/hip_runtime.h>

// ---------------------------------------------------------------------------
// GCN forward on gfx1250: WMMA f32 GEMMs (LDS-staged, branch-free inner loop)
// + L2-resident atomic SpMM + fused BN.  N=50000, F=128 hidden, 40 out.
// ---------------------------------------------------------------------------

typedef __attribute__((ext_vector_type(2))) float v2f;
typedef __attribute__((ext_vector_type(8))) float v8f;

#define LDSPAD 132  // 128 + 4: stride-132 column reads land on distinct LDS banks

// Y[rowBase : rowBase+16*TILES, :] = X[..., 0:128] @ W[0:128, 0:nColOut]
// NCP = padded column count (16 * number of waves), TILES = row tiles per block.
// Both X-slab and W are staged in LDS; W is stored K-pair-major so each lane's
// B fragment {W[k,col], W[k+1,col]} is a single aligned ds_load_b64.
template <int NCP, int TILES>
__global__ void gemm_k128_wmma(const float* __restrict__ X,
                               const float* __restrict__ W,
                               float* __restrict__ Y,
                               int nColOut, int ldW, int ldY) {
    constexpr int ROWS = TILES * 16;
    __shared__ float sX[ROWS * LDSPAD];
    __shared__ float sW[128 * NCP];  // [k/2][col][k&1]

    const int tid = threadIdx.x;
    const int rowBase = blockIdx.x * ROWS;

    // Stage ROWSx128 slab of X (contiguous), coalesced.
    const float* src = X + (size_t)rowBase * 128;
    for (int i = tid; i < ROWS * 128; i += blockDim.x)
        sX[(i >> 7) * LDSPAD + (i & 127)] = src[i];

    // Stage W with zero-padding for col >= nColOut; K-pair-major layout.
    for (int i = tid; i < 128 * NCP; i += blockDim.x) {
        int k = i / NCP;
        int c = i - k * NCP;
        float v = (c < nColOut) ? W[(size_t)k * ldW + c] : 0.0f;
        sW[(k >> 1) * (2 * NCP) + (c << 1) + (k & 1)] = v;
    }
    __syncthreads();

    const int lane = tid & 31;
    const int wv   = tid >> 5;
    const int m  = lane & 15;          // A: row M ; B/C: column N
    const int kh = (lane >> 4) << 1;   // lanes 0-15 -> K {0,1}; 16-31 -> K {2,3}
    const int col = (wv << 4) + m;

    v8f acc[TILES];
    #pragma unroll
    for (int t = 0; t < TILES; ++t) acc[t] = (v8f){};

    for (int k0 = 0; k0 < 128; k0 += 4) {
        const v2f b = *(const v2f*)&sW[((k0 + kh) >> 1) * (2 * NCP) + (col << 1)];
        #pragma unroll
        for (int t = 0; t < TILES; ++t) {
            const v2f a = *(const v2f*)&sX[(t * 16 + m) * LDSPAD + k0 + kh];
            // 8 args: (neg_a, A, neg_b, B, c_mod, C, reuse_a, reuse_b)
            acc[t] = __builtin_amdgcn_wmma_f32_16x16x4_f32(
                false, a, false, b, (short)0, acc[t], false, false);
        }
    }

    if (col < nColOut) {
        // C layout: VGPR r <-> M = r (lanes 0-15) / r+8 (lanes 16-31), N = lane%16
        const int mOff = (lane >> 4) << 3;
        #pragma unroll
        for (int t = 0; t < TILES; ++t) {
            float* dst = Y + (size_t)(rowBase + t * 16 + mOff) * ldY + col;
            #pragma unroll
            for (int r = 0; r < 8; ++r) dst[(size_t)r * ldY] = acc[t][r];
        }
    }
}

__global__ void zero_f32v4(float4* __restrict__ p, long long n4) {
    long long i = (long long)blockIdx.x * blockDim.x + threadIdx.x;
    if (i < n4) p[i] = make_float4(0.f, 0.f, 0.f, 0.f);
}

// F=128: one wave per edge. Edge indices are wave-uniform (-> scalar loads),
// gather is one b128 load per lane, 4 consecutive-address f32 atomics.
__global__ void spmm_scatter128_wave(const int* __restrict__ erow,
                                     const int* __restrict__ ecol,
                                     const float* __restrict__ ew,
                                     const float* __restrict__ H,
                                     float* __restrict__ out, int E) {
    const int e = blockIdx.x * (blockDim.x >> 5) + (threadIdx.x >> 5);
    if (e >= E) return;
    const int lane = threadIdx.x & 31;
    const int r = erow[e];
    const int c = ecol[e];
    const float wgt = ew[e];
    const float4 v = *(const float4*)&H[(size_t)c * 128 + (lane << 2)];
    float* dst = &out[(size_t)r * 128 + (lane << 2)];
    __hip_atomic_fetch_add(dst + 0, wgt * v.x, __ATOMIC_RELAXED, __HIP_MEMORY_SCOPE_AGENT);
    __hip_atomic_fetch_add(dst + 1, wgt * v.y, __ATOMIC_RELAXED, __HIP_MEMORY_SCOPE_AGENT);
    __hip_atomic_fetch_add(dst + 2, wgt * v.z, __ATOMIC_RELAXED, __HIP_MEMORY_SCOPE_AGENT);
    __hip_atomic_fetch_add(dst + 3, wgt * v.w, __ATOMIC_RELAXED, __HIP_MEMORY_SCOPE_AGENT);
}

// Generic element-per-thread scatter (used for F=40 final layer).
template <int F>
__global__ void spmm_scatter(const int* __restrict__ erow,
                             const int* __restrict__ ecol,
                             const float* __restrict__ ew,
                             const float* __restrict__ H,
                             float* __restrict__ out, int E) {
    long long idx = (long long)blockIdx.x * blockDim.x + threadIdx.x;
    if (idx >= (long long)E * F) return;
    int e = (int)(idx / F);
    int f = (int)(idx - (long long)e * F);
    float v = ew[e] * H[(size_t)ecol[e] * F + f];
    __hip_atomic_fetch_add(&out[(size_t)erow[e] * F + f], v,
                           __ATOMIC_RELAXED, __HIP_MEMORY_SCOPE_AGENT);
}

// Per-column sum and sum-of-squares partials -> atomics into stats[0:128|128:256]
__global__ void bn_stats(const float* __restrict__ A, float* __restrict__ stats,
                         int nRows, int rowsPerBlock) {
    const int f  = threadIdx.x;  // 128 threads = one column each
    const int r0 = blockIdx.x * rowsPerBlock;
    int r1 = r0 + rowsPerBlock;
    if (r1 > nRows) r1 = nRows;
    float s = 0.0f, s2 = 0.0f;
    for (int r = r0; r < r1; ++r) {
        float v = A[(size_t)r * 128 + f];
        s += v;
        s2 += v * v;
    }
    __hip_atomic_fetch_add(&stats[f], s, __ATOMIC_RELAXED, __HIP_MEMORY_SCOPE_AGENT);
    __hip_atomic_fetch_add(&stats[128 + f], s2, __ATOMIC_RELAXED, __HIP_MEMORY_SCOPE_AGENT);
}

// Y = relu((A - mean) * rsqrt(var + eps) * gamma + beta)
// The pre-BN conv bias shifts mean identically, so it cancels and is omitted.
__global__ void bn_relu_apply(const float* __restrict__ A,
                              const float* __restrict__ stats,
                              const float* __restrict__ g,
                              const float* __restrict__ be,
                              float* __restrict__ Y, long long n, float invN) {
    long long idx = (long long)blockIdx.x * blockDim.x + threadIdx.x;
    if (idx >= n) return;
    int f = (int)(idx & 127);
    float mean = stats[f] * invN;
    float var  = stats[128 + f] * invN - mean * mean;
    float v = (A[idx] - mean) * rsqrtf(var + 1e-5f) * g[f] + be[f];
    Y[idx] = v > 0.0f ? v : 0.0f;
}

__global__ void add_bias40(float* __restrict__ out, const float* __restrict__ b,
                           long long n) {
    long long idx = (long long)blockIdx.x * blockDim.x + threadIdx.x;
    if (idx >= n) return;
    int f = (int)(idx % 40);
    out[idx] += b[f];
}

// ---------------------------------------------------------------------------

extern "C" void kernel_launch(void* const* d_in, const int* in_sizes, int n_in,
                              void* d_out, int out_size, void* d_ws, size_t ws_size,
                              hipStream_t stream) {
    // setup_inputs() order:
    const float* x    = (const float*)d_in[0];   // [N,128]
    const int*   erow = (const int*)d_in[1];     // [E]
    const int*   ecol = (const int*)d_in[2];     // [E]
    const float* ew   = (const float*)d_in[3];   // [E]
    const float* W0   = (const float*)d_in[4];   // [128,128]
    // d_in[5] = b0 (cancels through BN)
    const float* g0   = (const float*)d_in[6];
    const float* be0  = (const float*)d_in[7];
    const float* W1   = (const float*)d_in[8];   // [128,128]
    // d_in[9] = b1 (cancels through BN)
    const float* g1   = (const float*)d_in[10];
    const float* be1  = (const float*)d_in[11];
    const float* W2   = (const float*)d_in[12];  // [128,40]
    const float* b2   = (const float*)d_in[13];  // [40]
    float* out = (float*)d_out;                  // [N,40]

    const int N = in_sizes[0] / 128;
    const int E = in_sizes[1];
    const float invN = 1.0f / (float)N;
    const long long nh = (long long)N * 128;     // hidden elements
    const long long no = (long long)N * 40;      // output elements

    // Workspace: two ping-pong [N,128] buffers + 256-float BN stats.
    float* H     = (float*)d_ws;       // buffer 0
    float* A     = H + nh;             // buffer 1
    float* stats = A + nh;             // 256 floats
    (void)ws_size; (void)n_in; (void)out_size;

    const int gemmBlocks = N / 80;               // 625 (TILES=5 -> 80 rows/block)
    const int ZB = 256;
    const int spmmBlocks128 = (E + 7) / 8;       // 8 edges (waves) per block
    const int spmmBlocks40  = (int)(((long long)E * 40 + ZB - 1) / ZB);
    const int zhBlocks = (int)((nh / 4 + ZB - 1) / ZB);
    const int zoBlocks = (int)((no / 4 + ZB - 1) / ZB);
    const int ehBlocks = (int)((nh + ZB - 1) / ZB);
    const int eoBlocks = (int)((no + ZB - 1) / ZB);
    const int rowsPerBlock = 128;
    const int statBlocks = (N + rowsPerBlock - 1) / rowsPerBlock;

    // ----- Layer 0: conv -> BN -> ReLU -----
    gemm_k128_wmma<128, 5><<<gemmBlocks, 256, 0, stream>>>(x, W0, H, 128, 128, 128);
    zero_f32v4<<<zhBlocks, ZB, 0, stream>>>((float4*)A, nh / 4);
    spmm_scatter128_wave<<<spmmBlocks128, 256, 0, stream>>>(erow, ecol, ew, H, A, E);
    zero_f32v4<<<1, 64, 0, stream>>>((float4*)stats, 64);
    bn_stats<<<statBlocks, 128, 0, stream>>>(A, stats, N, rowsPerBlock);
    bn_relu_apply<<<ehBlocks, ZB, 0, stream>>>(A, stats, g0, be0, H, nh, invN);

    // ----- Layer 1: conv -> BN -> ReLU -----
    gemm_k128_wmma<128, 5><<<gemmBlocks, 256, 0, stream>>>(H, W1, A, 128, 128, 128);
    zero_f32v4<<<zhBlocks, ZB, 0, stream>>>((float4*)H, nh / 4);
    spmm_scatter128_wave<<<spmmBlocks128, 256, 0, stream>>>(erow, ecol, ew, A, H, E);
    zero_f32v4<<<1, 64, 0, stream>>>((float4*)stats, 64);
    bn_stats<<<statBlocks, 128, 0, stream>>>(H, stats, N, rowsPerBlock);
    bn_relu_apply<<<ehBlocks, ZB, 0, stream>>>(H, stats, g1, be1, A, nh, invN);

    // ----- Final conv (128 -> 40), bias applied after aggregation -----
    float* Fbuf = H;  // [N,40] reuses buffer 0 (free after layer-1 BN)
    gemm_k128_wmma<48, 5><<<gemmBlocks, 96, 0, stream>>>(A, W2, Fbuf, 40, 40, 40);
    zero_f32v4<<<zoBlocks, ZB, 0, stream>>>((float4*)out, no / 4);
    spmm_scatter<40><<<spmmBlocks40, ZB, 0, stream>>>(erow, ecol, ew, Fbuf, out, E);
    add_bias40<<<eoBlocks, ZB, 0, stream>>>(out, b2, no);
}